// QueryAndGroup_57552561766925
// MI455X (gfx1250) — compile-verified
//
#include <hip/hip_runtime.h>

// ---- problem constants (match reference shapes) ----
#define BQ    2
#define NPTS  16384
#define MQ    4096
#define CF    64
#define NS    32
#define RAD2  1.0f

#define NF_STRIDE ((3 + CF) * NS)                    // 67*32 = 2144 floats per query row
#define NF_ELEMS  ((size_t)BQ * MQ * NF_STRIDE)      // new_features element count

typedef __attribute__((ext_vector_type(2))) float v2f;
typedef __attribute__((ext_vector_type(8))) float v8f;

// ---------------------------------------------------------------------------
// Kernel 1: ball query via V_WMMA_F32_16X16X4_F32.
// One wave (32 lanes) handles a tile of 16 queries; loops over points in
// chunks of 16. WMMA computes D = A*B + C = d^2 - R^2 for a 16x16 tile.
//   A (16x4):  row m = [qx qy qz 1]          (lanes 0-15: K0,K1; 16-31: K2,K3)
//   B (4x16):  col n = [-2px -2py -2pz |p|^2] (lanes 0-15: K0,K1; 16-31: K2,K3)
//   C (16x16): c[j] = |q_row|^2 - R^2        (rows j / j+8 per lane half)
// Ballot(D<0) bits[15:0] = row j cols 0..15, bits[31:16] = row j+8 cols 0..15.
// ---------------------------------------------------------------------------
__global__ __launch_bounds__(32)
void ball_query_wmma_kernel(const float* __restrict__ xyz,
                            const float* __restrict__ new_xyz,
                            int* __restrict__ out_idx,
                            int* __restrict__ out_cnt) {
  const int  lane    = threadIdx.x & 31;
  const bool hiHalf  = lane >= 16;
  const int  r       = lane & 15;
  const int  q_base  = blockIdx.x * 16;
  const int  batch   = q_base / MQ;            // 16 | MQ, tiles never straddle batches

  // ---- A matrix + C accumulator (per 16-query tile, loaded once) ----
  const int   qrow = q_base + r;
  const float qx = new_xyz[qrow * 3 + 0];
  const float qy = new_xyz[qrow * 3 + 1];
  const float qz = new_xyz[qrow * 3 + 2];
  const float qq = qx * qx + qy * qy + qz * qz;

  v2f a;
  a[0] = hiHalf ? qz   : qx;    // K2 : K0
  a[1] = hiHalf ? 1.0f : qy;    // K3 : K1

  v8f cacc;
#pragma unroll
  for (int j = 0; j < 8; ++j)
    cacc[j] = __shfl(qq, j + (hiHalf ? 8 : 0), 32) - RAD2;

  int cnt = 0;
  int first = 0;
  const float* pbase = xyz + (size_t)batch * NPTS * 3;

  for (int base = 0; base < NPTS; base += 16) {
    // ---- B matrix: 16 points of this chunk ----
    const float* p = pbase + (size_t)(base + r) * 3;
    const float px = p[0], py = p[1], pz = p[2];
    if (base + 16 < NPTS)
      __builtin_prefetch(pbase + (size_t)(base + 16 + r) * 3, 0, 1);
    const float pp = px * px + py * py + pz * pz;

    v2f b;
    b[0] = hiHalf ? (-2.0f * pz) : (-2.0f * px);  // K2 : K0
    b[1] = hiHalf ? pp           : (-2.0f * py);  // K3 : K1

    // D = A*B + C  ->  d^2 - R^2 for (query row, point col)
    v8f d = __builtin_amdgcn_wmma_f32_16x16x4_f32(
        false, a, false, b, (short)0, cacc, false, false);

    unsigned mm[8];
#pragma unroll
    for (int j = 0; j < 8; ++j)
      mm[j] = (unsigned)__ballot(d[j] < 0.0f);

    // Lane r (<16) owns query row r: extract its 16-bit column mask.
    if (lane < 16 && cnt < NS) {
      unsigned rowmask = 0;
#pragma unroll
      for (int j = 0; j < 8; ++j) {
        rowmask = (lane == j)     ? (mm[j] & 0xFFFFu) : rowmask;
        rowmask = (lane == j + 8) ? (mm[j] >> 16)     : rowmask;
      }
      while (rowmask && cnt < NS) {
        const int col = __builtin_ctz(rowmask);
        rowmask &= rowmask - 1u;
        const int pidx = base + col;             // batch-local index (reference semantics)
        if (cnt == 0) first = pidx;
        out_idx[(size_t)(q_base + lane) * NS + cnt] = pidx;
        ++cnt;
      }
    }

    const bool done = hiHalf || (cnt >= NS);
    if (__ballot(!done) == 0ull) break;          // wave-uniform early exit
  }

  if (lane < 16) {
    const int q    = q_base + lane;
    const int fill = (cnt > 0) ? first : 0;      // pad with first hit; 0 if empty ball
    for (int s = cnt; s < NS; ++s)
      out_idx[(size_t)q * NS + s] = fill;
    out_cnt[q] = cnt;
  }
}

// ---------------------------------------------------------------------------
// Kernel 2: grouping / gather. Block = (32 samples, 8 queries).
// Output layout (B*M, 3+C, NS): channel-major over samples -> for a fixed
// channel, lanes s=0..31 store 128B contiguous (fully coalesced).
// ---------------------------------------------------------------------------
__global__ __launch_bounds__(256)
void group_gather_kernel(const float* __restrict__ xyz,
                         const float* __restrict__ new_xyz,
                         const float* __restrict__ feat,
                         const int* __restrict__ idx,
                         const int* __restrict__ cnts,
                         float* __restrict__ out_nf) {
  const int s = threadIdx.x;                                  // sample 0..31
  const int q = blockIdx.x * blockDim.y + threadIdx.y;        // global query row
  const int batch = q / MQ;
  const int id    = idx[(size_t)q * NS + s];                  // 0 when empty
  const bool empty = (cnts[q] == 0);

  const float* p = xyz + (size_t)(batch * NPTS + id) * 3;
  const float rx = empty ? 0.0f : (p[0] - new_xyz[q * 3 + 0]);
  const float ry = empty ? 0.0f : (p[1] - new_xyz[q * 3 + 1]);
  const float rz = empty ? 0.0f : (p[2] - new_xyz[q * 3 + 2]);

  float* o = out_nf + (size_t)q * NF_STRIDE;
  o[0 * NS + s] = rx;
  o[1 * NS + s] = ry;
  o[2 * NS + s] = rz;

  const float4* f = (const float4*)(feat + (size_t)(batch * NPTS + id) * CF);
#pragma unroll
  for (int c4 = 0; c4 < CF / 4; ++c4) {
    float4 v = f[c4];                                         // L2-resident gather
    o[(3 + 4 * c4 + 0) * NS + s] = empty ? 0.0f : v.x;
    o[(3 + 4 * c4 + 1) * NS + s] = empty ? 0.0f : v.y;
    o[(3 + 4 * c4 + 2) * NS + s] = empty ? 0.0f : v.z;
    o[(3 + 4 * c4 + 3) * NS + s] = empty ? 0.0f : v.w;
  }
}

// ---------------------------------------------------------------------------
extern "C" void kernel_launch(void* const* d_in, const int* in_sizes, int n_in,
                              void* d_out, int out_size, void* d_ws, size_t ws_size,
                              hipStream_t stream) {
  // setup_inputs order: xyz, xyz_batch_cnt, new_xyz, new_xyz_batch_cnt, features
  const float* xyz     = (const float*)d_in[0];
  const float* new_xyz = (const float*)d_in[2];
  const float* feat    = (const float*)d_in[4];

  float* out_nf  = (float*)d_out;
  int*   out_idx = (int*)(out_nf + NF_ELEMS);   // second tuple output, flat-concatenated
  int*   cnts    = (int*)d_ws;                  // per-query hit count (B*M ints)

  // Phase 1: WMMA ball query (one wave per 16 queries).
  ball_query_wmma_kernel<<<dim3((BQ * MQ) / 16), dim3(32), 0, stream>>>(
      xyz, new_xyz, out_idx, cnts);

  // Phase 2: bandwidth-bound grouping.
  group_gather_kernel<<<dim3((BQ * MQ) / 8), dim3(32, 8), 0, stream>>>(
      xyz, new_xyz, feat, out_idx, cnts, out_nf);
}